// RadarFlowLoss_41042707481159
// MI455X (gfx1250) — compile-verified
//
#include <hip/hip_runtime.h>
#include <math.h>

// ---------------------------------------------------------------------------
// RadarFlowLoss for MI455X (gfx1250, wave32).
// - Gram tiles via V_WMMA_F32_16X16X4_F32 (K=3 padded to 4).
// - pc2 / pc1 batch slices staged to LDS with GLOBAL_LOAD_ASYNC_TO_LDS_B128
//   (ASYNCcnt + s_wait_asynccnt), making the 512-iteration inner loops pure
//   branchless ds_load + VALU/WMMA.
// Problem is compute bound (~786KB input vs ~400M tile elements), so the
// matrix pipe carries the inner products and VALU carries exp/min/top-k.
// ---------------------------------------------------------------------------

typedef __attribute__((ext_vector_type(2))) float v2f;
typedef __attribute__((ext_vector_type(8))) float v8f;
typedef int v4i __attribute__((vector_size(16)));

#define BB 2
#define NNP 8192
#define NT (NNP / 16)          // 512 16-wide tiles per axis
#define ZETA 0.005f
#define CH_LB 0.01f
#define INV25 0.4f             // 1/(2.5*bw), bw=1
#define FINF 3.402823466e38f

#define AS1 __attribute__((address_space(1)))
#define AS3 __attribute__((address_space(3)))

__device__ __forceinline__ v8f wmma_k4(v2f a, v2f b, v8f c) {
  // D = A(16x4,f32) * B(4x16,f32) + C(16x16,f32)   -> v_wmma_f32_16x16x4_f32
  return __builtin_amdgcn_wmma_f32_16x16x4_f32(false, a, false, b, (short)0, c,
                                               false, false);
}

// async copy: 16B per lane, global -> LDS (tracked by ASYNCcnt)
__device__ __forceinline__ void async_b128(const float* g, float* l) {
  __builtin_amdgcn_global_load_async_to_lds_b128((AS1 v4i*)(void*)(size_t)(const void*)g,
                                                 (AS3 v4i*)l, 0, 0);
}
__device__ __forceinline__ void wait_async0() {
  asm volatile("s_wait_asynccnt 0" ::: "memory");
}
__device__ __forceinline__ void wave_lds_fence() {
  asm volatile("s_wait_dscnt 0" ::: "memory");
}

// order-preserving float->uint map for atomicMin on possibly-tiny-negative sqrd
__device__ __forceinline__ unsigned fkey(float f) {
  unsigned u = __float_as_uint(f);
  return (u & 0x80000000u) ? ~u : (u | 0x80000000u);
}
__device__ __forceinline__ float funkey(unsigned k) {
  unsigned u = (k & 0x80000000u) ? (k & 0x7FFFFFFFu) : ~k;
  return __uint_as_float(u);
}
__device__ __forceinline__ float wsum32(float v) {
#pragma unroll
  for (int o = 16; o > 0; o >>= 1) v += __shfl_xor(v, o, 32);
  return v;
}

// ---------------------------------------------------------------- init ------
__global__ void init_kernel(float* dens2, unsigned* dist2u, float* scal) {
  int i = blockIdx.x * blockDim.x + threadIdx.x;
  if (i < BB * NNP) { dens2[i] = 0.f; dist2u[i] = 0xFFFFFFFFu; }
  if (i < 8) scal[i] = 0.f;
}

// ------------------------------------------------------------- norms --------
__global__ void norms_kernel(const float* __restrict__ pc1,
                             const float* __restrict__ pc2,
                             const float* __restrict__ pf,
                             float* xn1, float* wn, float* yn2) {
  int idx = blockIdx.x * blockDim.x + threadIdx.x;
  if (idx >= BB * NNP) return;
  int b = idx / NNP, n = idx % NNP;
  const float* p1 = pc1 + (size_t)b * 3 * NNP;
  const float* p2 = pc2 + (size_t)b * 3 * NNP;
  const float* f  = pf  + (size_t)b * 3 * NNP;
  float x0 = p1[n], x1 = p1[NNP + n], x2 = p1[2 * NNP + n];
  float y0 = p2[n], y1 = p2[NNP + n], y2 = p2[2 * NNP + n];
  float w0 = x0 + f[n], w1 = x1 + f[NNP + n], w2 = x2 + f[2 * NNP + n];
  xn1[idx] = x0 * x0 + x1 * x1 + x2 * x2;
  yn2[idx] = y0 * y0 + y1 * y1 + y2 * y2;
  wn[idx]  = w0 * w0 + w1 * w1 + w2 * w2;
}

// ---------------------------------------- fused density + chamfer ----------
// 8 waves/block, one wave per 16-row i-tile; each block async-stages the
// whole pc2 batch slice (3 channels) + yn2 into LDS, then loops all j-tiles
// with two WMMAs per tile (Gram(pc1,pc2) for density, Gram(warp,pc2) for
// chamfer). Row quantities live in registers; column quantities are merged
// across wave halves with a lane-16 shuffle, then pushed with atomics.
__global__ __launch_bounds__(256) void chamfer_kernel(
    const float* __restrict__ pc1, const float* __restrict__ pc2,
    const float* __restrict__ pf, const float* __restrict__ xn1,
    const float* __restrict__ wn, const float* __restrict__ yn2,
    float* dens1, float* dens2, float* dist1, unsigned* dist2u) {
  __shared__ float ldsP2[3 * NNP];   // 96KB: pc2 slice, channel-major
  __shared__ float ldsYN[NNP];       // 32KB: |pc2|^2
  int b = (blockIdx.x * 8) / NT;     // block-uniform batch index
  int wave = blockIdx.x * 8 + (threadIdx.x >> 5);
  int lane = threadIdx.x & 31;
  int ibase = (wave % NT) * 16;
  const float* p1 = pc1 + (size_t)b * 3 * NNP;
  const float* p2 = pc2 + (size_t)b * 3 * NNP;
  const float* f  = pf  + (size_t)b * 3 * NNP;

  // ---- async stage pc2 slice + norms to LDS ----
  for (int t = threadIdx.x; t < (3 * NNP) / 4; t += 256)
    async_b128(p2 + 4 * t, ldsP2 + 4 * t);
  for (int t = threadIdx.x; t < NNP / 4; t += 256)
    async_b128(yn2 + (size_t)b * NNP + 4 * t, ldsYN + 4 * t);

  int r = lane & 15;
  bool hi = lane >= 16;
  int mb = hi ? 8 : 0;
  int ch0 = hi ? 2 * NNP : 0;        // B-fragment channel base per half-wave

  // A fragments (16x4, K=3 padded): lanes 0-15 -> (K0,K1), lanes 16-31 -> (K2,0)
  v2f A1, AW;
  {
    int g = ibase + r;
    if (!hi) {
      float a0 = p1[g], a1 = p1[NNP + g];
      A1 = (v2f){a0, a1};
      AW = (v2f){a0 + f[g], a1 + f[NNP + g]};
    } else {
      float a0 = p1[2 * NNP + g];
      A1 = (v2f){a0, 0.f};
      AW = (v2f){a0 + f[2 * NNP + g], 0.f};
    }
  }
  float rn1[8], rnw[8], rowsum[8], rowmin[8];
#pragma unroll
  for (int v = 0; v < 8; ++v) {
    rn1[v] = xn1[b * NNP + ibase + mb + v];
    rnw[v] = wn[b * NNP + ibase + mb + v];
    rowsum[v] = 0.f;
    rowmin[v] = FINF;
  }

  wait_async0();
  __syncthreads();

  for (int jt = 0; jt < NT; ++jt) {
    int jbase = jt * 16;
    int g = jbase + r;
    // branchless B fragment from LDS
    float b0 = ldsP2[ch0 + g];
    float b1 = ldsP2[NNP + g];
    b1 = hi ? 0.f : b1;
    v2f Bf = (v2f){b0, b1};

    v8f c0 = {0.f, 0.f, 0.f, 0.f, 0.f, 0.f, 0.f, 0.f};
    v8f g1 = wmma_k4(A1, Bf, c0);   // pc1 . pc2
    v8f g2 = wmma_k4(AW, Bf, c0);   // warp . pc2

    float cn = ldsYN[g];
    float colsum = 0.f, colmin = FINF;
#pragma unroll
    for (int v = 0; v < 8; ++v) {
      float d12 = rn1[v] + cn - 2.f * g1[v];
      float gg = __expf(-0.5f * d12) * INV25;
      rowsum[v] += gg;
      colsum += gg;
      float dw = rnw[v] + cn - 2.f * g2[v];
      rowmin[v] = fminf(rowmin[v], dw);
      colmin = fminf(colmin, dw);
    }
    // merge the two half-wave contributions for column j = jbase + r
    colsum += __shfl_xor(colsum, 16, 32);
    colmin = fminf(colmin, __shfl_xor(colmin, 16, 32));
    if (!hi) {
      atomicAdd(&dens2[b * NNP + g], colsum);
      atomicMin(&dist2u[b * NNP + g], fkey(colmin));
    }
  }
  // reduce row quantities across the 16 lanes of each half-wave
#pragma unroll
  for (int v = 0; v < 8; ++v) {
    float s = rowsum[v], m = rowmin[v];
#pragma unroll
    for (int o = 1; o < 16; o <<= 1) {
      s += __shfl_xor(s, o, 16);
      m = fminf(m, __shfl_xor(m, o, 16));
    }
    if (r == 0) {
      dens1[b * NNP + ibase + mb + v] = s;
      dist1[b * NNP + ibase + mb + v] = m;
    }
  }
}

// ----------------------------------------------- chamfer finalize ----------
__global__ void chamfer_fin(const float* dens1, const float* dens2,
                            const float* dist1, const unsigned* dist2u,
                            float* scal) {
  int idx = blockIdx.x * blockDim.x + threadIdx.x;
  float s1 = 0.f, s2 = 0.f;
  if (idx < BB * NNP) {
    float m1 = (dens1[idx] * (1.f / NNP) > ZETA) ? 1.f : 0.f;
    float m2 = (dens2[idx] * (1.f / NNP) > ZETA) ? 1.f : 0.f;
    s1 = fmaxf(dist1[idx] - CH_LB, 0.f) * m1;
    s2 = fmaxf(funkey(dist2u[idx]) - CH_LB, 0.f) * m2;
  }
  s1 = wsum32(s1);
  s2 = wsum32(s2);
  if ((threadIdx.x & 31) == 0) {
    atomicAdd(&scal[0], s1);
    atomicAdd(&scal[1], s2);
  }
}

// ------------------------------------------------ 8-NN via WMMA ------------
// 8 waves/block, one wave per 16-row i-tile; pc1 slice + norms async-staged
// to LDS. Each wave computes WMMA distance tiles, stages them through a
// private LDS tile (wave-local s_wait_dscnt fence, LDS is in-order per wave),
// and lanes 0-15 maintain per-row top-9 insertion lists (self forced to
// -inf, dropped as entry 0).
__global__ __launch_bounds__(256) void knn_kernel(const float* __restrict__ pc1,
                                                  const float* __restrict__ xn1,
                                                  float* nbD, int* nbI) {
  __shared__ float ldsP1[3 * NNP];   // 96KB
  __shared__ float ldsXN[NNP];       // 32KB
  __shared__ float tile[8][16][17];  // per-wave distance tile
  int b = (blockIdx.x * 8) / NT;
  int wave = blockIdx.x * 8 + (threadIdx.x >> 5);
  int wid = (threadIdx.x >> 5) & 7;
  int lane = threadIdx.x & 31;
  int ibase = (wave % NT) * 16;
  const float* p1 = pc1 + (size_t)b * 3 * NNP;

  for (int t = threadIdx.x; t < (3 * NNP) / 4; t += 256)
    async_b128(p1 + 4 * t, ldsP1 + 4 * t);
  for (int t = threadIdx.x; t < NNP / 4; t += 256)
    async_b128(xn1 + (size_t)b * NNP + 4 * t, ldsXN + 4 * t);

  int r = lane & 15;
  bool hi = lane >= 16;
  int mb = hi ? 8 : 0;
  int ch0 = hi ? 2 * NNP : 0;

  v2f A;
  {
    int g = ibase + r;
    A = (!hi) ? (v2f){p1[g], p1[NNP + g]} : (v2f){p1[2 * NNP + g], 0.f};
  }
  float rn[8];
#pragma unroll
  for (int v = 0; v < 8; ++v) rn[v] = xn1[b * NNP + ibase + mb + v];

  float nd[9];
  int ni[9];
#pragma unroll
  for (int k = 0; k < 9; ++k) { nd[k] = FINF; ni[k] = 0; }

  wait_async0();
  __syncthreads();

  for (int jt = 0; jt < NT; ++jt) {
    int jbase = jt * 16;
    int g = jbase + r;
    float b0 = ldsP1[ch0 + g];
    float b1 = ldsP1[NNP + g];
    b1 = hi ? 0.f : b1;
    v2f Bf = (v2f){b0, b1};
    v8f c0 = {0.f, 0.f, 0.f, 0.f, 0.f, 0.f, 0.f, 0.f};
    v8f gm = wmma_k4(A, Bf, c0);
    float cn = ldsXN[g];
#pragma unroll
    for (int v = 0; v < 8; ++v) {
      float d = rn[v] + cn - 2.f * gm[v];
      if (ibase + mb + v == g) d = -FINF;  // self -> always entry 0, dropped
      tile[wid][mb + v][r] = d;
    }
    wave_lds_fence();                 // LDS in-order per wave; fence compiler
    if (lane < 16) {
      for (int n = 0; n < 16; ++n) {
        float d = tile[wid][lane][n];
        if (d < nd[8]) {
          int p = 8;
          while (p > 0 && nd[p - 1] > d) {
            nd[p] = nd[p - 1]; ni[p] = ni[p - 1]; --p;
          }
          nd[p] = d; ni[p] = jbase + n;
        }
      }
    }
    wave_lds_fence();
  }
  if (lane < 16) {
    size_t base = ((size_t)b * NNP + ibase + lane) * 8;
#pragma unroll
    for (int k = 1; k < 9; ++k) { nbD[base + k - 1] = nd[k]; nbI[base + k - 1] = ni[k]; }
  }
}

// --------------------------------------- smoothness softmax passes ---------
__global__ void ss_pass1(const float* __restrict__ nbD, float* scal) {
  int idx = blockIdx.x * blockDim.x + threadIdx.x;  // BB*NNP*8, batch-aligned
  int b = idx / (NNP * 8);
  float d = fmaxf(nbD[idx], 0.f);
  float e = __expf(__expf(-2.f * d));               // exp(exp(-d/alpha)), alpha=0.5
  e = wsum32(e);
  if ((threadIdx.x & 31) == 0) atomicAdd(&scal[2 + b], e);
}

__global__ void ss_pass2(const float* __restrict__ nbD,
                         const int* __restrict__ nbI,
                         const float* __restrict__ pf, float* scal) {
  int idx = blockIdx.x * blockDim.x + threadIdx.x;  // BB*NNP
  int b = idx / NNP, i = idx % NNP;
  const float* f = pf + (size_t)b * 3 * NNP;
  float f0 = f[i], f1 = f[NNP + i], f2 = f[2 * NNP + i];
  float invS = 1.f / scal[2 + b];
  float acc = 0.f;
#pragma unroll
  for (int k = 0; k < 8; ++k) {
    float d = fmaxf(nbD[(size_t)idx * 8 + k], 0.f);
    int j = nbI[(size_t)idx * 8 + k];
    float w = __expf(__expf(-2.f * d)) * invS;
    float d0 = f[j] - f0, d1 = f[NNP + j] - f1, d2 = f[2 * NNP + j] - f2;
    acc += w * sqrtf(d0 * d0 + d1 * d1 + d2 * d2);
  }
  acc *= (float)NNP;
  acc = wsum32(acc);
  if ((threadIdx.x & 31) == 0) atomicAdd(&scal[4], acc);
}

// --------------------------------------------- radial displacement ---------
__global__ void rd_kernel(const float* __restrict__ pc1,
                          const float* __restrict__ pf,
                          const float* __restrict__ vel1, float* scal) {
  int idx = blockIdx.x * blockDim.x + threadIdx.x;  // BB*NNP
  int b = idx / NNP, n = idx % NNP;
  const float* p1 = pc1 + (size_t)b * 3 * NNP;
  const float* f  = pf  + (size_t)b * 3 * NNP;
  float x0 = p1[n], x1 = p1[NNP + n], x2 = p1[2 * NNP + n];
  float dot = f[n] * x0 + f[NNP + n] * x1 + f[2 * NNP + n] * x2;
  float fr = dot * __frsqrt_rn(x0 * x0 + x1 * x1 + x2 * x2);
  float a = fabsf(vel1[idx] * 0.1f - fr);
  a = wsum32(a);
  if ((threadIdx.x & 31) == 0) atomicAdd(&scal[5], a);
}

// -------------------------------------------------------- combine ----------
__global__ void combine_kernel(const float* scal, float* out) {
  if (threadIdx.x == 0 && blockIdx.x == 0) {
    float inv = 1.f / (float)(BB * NNP);
    out[0] = (scal[0] + scal[1] + scal[4] + scal[5]) * inv;
  }
}

// ---------------------------------------------------------------------------
extern "C" void kernel_launch(void* const* d_in, const int* in_sizes, int n_in,
                              void* d_out, int out_size, void* d_ws,
                              size_t ws_size, hipStream_t stream) {
  const float* pc1  = (const float*)d_in[0];
  const float* pc2  = (const float*)d_in[1];
  const float* pf   = (const float*)d_in[2];
  const float* vel1 = (const float*)d_in[3];
  float* out = (float*)d_out;

  const size_t BN = (size_t)BB * NNP;
  char* w = (char*)d_ws;
  float*    xn1    = (float*)w;             w += BN * 4;
  float*    wn     = (float*)w;             w += BN * 4;
  float*    yn2    = (float*)w;             w += BN * 4;
  float*    dens1  = (float*)w;             w += BN * 4;
  float*    dens2  = (float*)w;             w += BN * 4;
  float*    dist1  = (float*)w;             w += BN * 4;
  unsigned* dist2u = (unsigned*)w;          w += BN * 4;
  float*    nbD    = (float*)w;             w += BN * 8 * 4;
  int*      nbI    = (int*)w;               w += BN * 8 * 4;
  float*    scal   = (float*)w;

  init_kernel<<<(int)((BN + 255) / 256), 256, 0, stream>>>(dens2, dist2u, scal);
  norms_kernel<<<(int)(BN / 256), 256, 0, stream>>>(pc1, pc2, pf, xn1, wn, yn2);
  chamfer_kernel<<<(BB * NT) / 8, 256, 0, stream>>>(pc1, pc2, pf, xn1, wn, yn2,
                                                    dens1, dens2, dist1, dist2u);
  chamfer_fin<<<(int)(BN / 256), 256, 0, stream>>>(dens1, dens2, dist1, dist2u, scal);
  knn_kernel<<<(BB * NT) / 8, 256, 0, stream>>>(pc1, xn1, nbD, nbI);
  ss_pass1<<<(int)(BN * 8 / 256), 256, 0, stream>>>(nbD, scal);
  ss_pass2<<<(int)(BN / 256), 256, 0, stream>>>(nbD, nbI, pf, scal);
  rd_kernel<<<(int)(BN / 256), 256, 0, stream>>>(pc1, pf, vel1, scal);
  combine_kernel<<<1, 32, 0, stream>>>(scal, out);
}